// GNNPolicy_20332375179288
// MI455X (gfx1250) — compile-verified
//
#include <hip/hip_runtime.h>
#include <hip/hip_bf16.h>

// ---------------------------------------------------------------------------
// GraphSAGE 2-layer + head, fp32, CDNA5 (gfx1250).
//   h1 = relu(mean_agg(x) @ W1l + b1 + x @ W1r)
//   h2 = relu(mean_agg(h1) @ W2l + b2 + h1 @ W2r)
//   out = h2 @ Wa + ba
//
// Pipeline:
//   CSR build (int atomics only) -> wave-per-node mean aggregation (no float
//   atomics, L2-resident gathers) -> fp32 WMMA GEMMs (V_WMMA_F32_16X16X4_F32)
//   with pair-packed weights (b64 B-fragment loads) -> fused layer2+head with
//   head weights staged into LDS by the Tensor Data Mover (tensor_load_to_lds,
//   s_wait_tensorcnt), overlapped with the layer-2 WMMA loop.
// ---------------------------------------------------------------------------

#define N_NODES 50000
#define N_EDGES 800000
#define IN_C    128
#define HID_C   256
#define OUT_C   64

typedef __attribute__((ext_vector_type(2))) float v2f;
typedef __attribute__((ext_vector_type(4))) float v4f;
typedef __attribute__((ext_vector_type(8))) float v8f;
typedef __attribute__((ext_vector_type(4))) unsigned int v4u;
typedef __attribute__((ext_vector_type(4))) int v4i;
typedef __attribute__((ext_vector_type(8))) int v8i;

// ---------------------------------------------------------------------------
// CSR construction
// ---------------------------------------------------------------------------
__global__ void zero_i32_kernel(int* __restrict__ p, int n) {
    int i = blockIdx.x * blockDim.x + threadIdx.x;
    if (i < n) p[i] = 0;
}

__global__ void hist_kernel(const int* __restrict__ ei, int* __restrict__ deg) {
    int e = blockIdx.x * blockDim.x + threadIdx.x;
    if (e >= N_EDGES) return;
    atomicAdd(&deg[ei[N_EDGES + e]], 1);
}

// Single-block chunked exclusive scan: deg -> offs (+ cursor copy).
__global__ __launch_bounds__(1024) void scan_kernel(const int* __restrict__ deg,
                                                    int* __restrict__ offs,
                                                    int* __restrict__ cursor) {
    __shared__ int s[1024];
    __shared__ int carry_s;
    const int tid = threadIdx.x;
    if (tid == 0) carry_s = 0;
    __syncthreads();
    for (int base = 0; base < N_NODES; base += 1024) {
        const int i = base + tid;
        const int v = (i < N_NODES) ? deg[i] : 0;
        s[tid] = v;
        __syncthreads();
        // Hillis-Steele inclusive scan
        for (int off = 1; off < 1024; off <<= 1) {
            int t = (tid >= off) ? s[tid - off] : 0;
            __syncthreads();
            s[tid] += t;
            __syncthreads();
        }
        const int carry = carry_s;
        const int exc = s[tid] - v + carry;
        if (i < N_NODES) { offs[i] = exc; cursor[i] = exc; }
        __syncthreads();
        if (tid == 1023) carry_s = carry + s[1023];
        __syncthreads();
    }
    if (tid == 0) offs[N_NODES] = carry_s;
}

__global__ void scatter_edges_kernel(const int* __restrict__ ei,
                                     int* __restrict__ cursor,
                                     int* __restrict__ csr_src) {
    int e = blockIdx.x * blockDim.x + threadIdx.x;
    if (e >= N_EDGES) return;
    int dst = ei[N_EDGES + e];
    int p = atomicAdd(&cursor[dst], 1);
    csr_src[p] = ei[e];
}

// ---------------------------------------------------------------------------
// Wave-per-node mean aggregation: agg[n] = (1/max(deg,1)) * sum_{src in N(n)} feat[src]
// ---------------------------------------------------------------------------
template <int C>
__global__ void aggregate_kernel(const float* __restrict__ feat,
                                 const int* __restrict__ offs,
                                 const int* __restrict__ csr_src,
                                 float* __restrict__ agg) {
    const int wid  = (blockIdx.x * blockDim.x + threadIdx.x) >> 5;  // node id
    const int lane = threadIdx.x & 31;
    if (wid >= N_NODES) return;
    const int b = offs[wid];
    const int e = offs[wid + 1];
    v4f a0 = {}, a1 = {};
    for (int j = b; j < e; ++j) {
        const int src = csr_src[j];                 // uniform across wave
        const v4f* row = (const v4f*)(feat + (size_t)src * C);
        a0 += row[lane];
        if (C == 256) a1 += row[lane + 32];
    }
    const float inv = 1.0f / fmaxf((float)(e - b), 1.0f);
    a0 *= inv;
    v4f* orow = (v4f*)(agg + (size_t)wid * C);
    orow[lane] = a0;
    if (C == 256) { a1 *= inv; orow[lane + 32] = a1; }
}

// ---------------------------------------------------------------------------
// Pair-pack weights: Wp[(k/2)*(2N) + col*2 + (k&1)] = W[k*N + col]
// so a B fragment {W[k][col], W[k+1][col]} is one aligned b64 load.
// ---------------------------------------------------------------------------
__global__ void pack_pairs_kernel(const float* __restrict__ w,
                                  float* __restrict__ wp, int K, int N) {
    int i = blockIdx.x * blockDim.x + threadIdx.x;
    if (i >= K * N) return;
    int k = i / N, col = i - k * N;
    wp[(size_t)(k >> 1) * (2 * N) + col * 2 + (k & 1)] = w[i];
}

// ---------------------------------------------------------------------------
// Layer 1: h1 = relu(agg @ W1l + b1 + x @ W1r). 128 thr (4 waves), 16 rows,
// wave owns 4 N-tiles of 16. Weights pre-packed in pairs.
// ---------------------------------------------------------------------------
__global__ __launch_bounds__(128) void layer1_kernel(
        const float* __restrict__ x, const float* __restrict__ agg,
        const float* __restrict__ W1lp, const float* __restrict__ b1,
        const float* __restrict__ W1rp, float* __restrict__ h1) {
    const int rowbase = blockIdx.x * 16;
    const int wave = threadIdx.x >> 5;
    const int lane = threadIdx.x & 31;
    const int half = lane >> 4;
    const int l16  = lane & 15;
    const int arow = rowbase + l16;

    const float* xr = x   + (size_t)arow * IN_C;
    const float* ar = agg + (size_t)arow * IN_C;

    v8f acc[4] = {};
    const int nt0 = wave * 4;
    for (int kb = 0; kb < IN_C; kb += 4) {
        const int k  = kb + half * 2;
        const int kp = k >> 1;                       // pair row in packed W
        v2f aagg = *(const v2f*)(ar + k);
        v2f ax   = *(const v2f*)(xr + k);
#pragma unroll
        for (int t = 0; t < 4; ++t) {
            const int col = (nt0 + t) * 16 + l16;
            v2f bl = *(const v2f*)(W1lp + (size_t)kp * (2 * HID_C) + col * 2);
            v2f br = *(const v2f*)(W1rp + (size_t)kp * (2 * HID_C) + col * 2);
            acc[t] = __builtin_amdgcn_wmma_f32_16x16x4_f32(
                false, aagg, false, bl, (short)0, acc[t], false, false);
            acc[t] = __builtin_amdgcn_wmma_f32_16x16x4_f32(
                false, ax, false, br, (short)0, acc[t], false, false);
        }
    }
#pragma unroll
    for (int t = 0; t < 4; ++t) {
        const int col = (nt0 + t) * 16 + l16;
        const float bias = b1[col];
#pragma unroll
        for (int g = 0; g < 8; ++g) {
            const int r = rowbase + g + half * 8;
            h1[(size_t)r * HID_C + col] = fmaxf(acc[t][g] + bias, 0.0f);
        }
    }
}

// ---------------------------------------------------------------------------
// Layer 2 + head (fused). Head weights (packed Wa, 64 KB) staged into LDS by
// the Tensor Data Mover at kernel entry, overlapped with the layer-2 WMMA loop.
// ---------------------------------------------------------------------------
__global__ __launch_bounds__(128) void layer2_head_kernel(
        const float* __restrict__ h1, const float* __restrict__ agg,
        const float* __restrict__ W2lp, const float* __restrict__ b2,
        const float* __restrict__ W2rp, const float* __restrict__ Wap,
        const float* __restrict__ ba, float* __restrict__ out) {
    __shared__ float h2s[16][HID_C + 4];     // pad kills bank conflicts
    __shared__ float Wa_s[(HID_C / 2) * (2 * OUT_C)];   // 16384 floats = 64 KB

    const int rowbase = blockIdx.x * 16;
    const int wave = threadIdx.x >> 5;
    const int lane = threadIdx.x & 31;
    const int half = lane >> 4;
    const int l16  = lane & 15;
    const int arow = rowbase + l16;

    // --- TDM: DMA packed Wa (contiguous 16384 f32) into LDS, wave 0 only.
    // D# per CDNA5 ISA 8.3/8.4: group0 {count=1, lds_addr, global_addr, type=2},
    // group1 {data_size=4B, tensor_dim0=16384, tensor_dim1=1, tile_dim0=16384,
    // tile_dim1=1, dim0_stride=16384}. Groups 2/3 zero (<=2D tensor).
    if (threadIdx.x < 32) {
        unsigned lds_addr = (unsigned)(size_t)&Wa_s[0];
        unsigned long long ga = (unsigned long long)(size_t)Wap;
        v4u g0;
        g0[0] = 1u;                                        // count=1
        g0[1] = lds_addr;                                  // lds_addr (bytes)
        g0[2] = (unsigned)(ga & 0xFFFFFFFFu);              // global_addr[31:0]
        g0[3] = (unsigned)((ga >> 32) & 0x01FFFFFFu) | (2u << 30);  // addr[56:32] | type=2
        v8i g1 = {};
        g1[0] = 2 << 16;                                   // data_size = 4 bytes
        g1[1] = 16384 << 16;                               // tensor_dim0[15:0]
        g1[2] = 1 << 16;                                   // tensor_dim1 = 1
        g1[3] = 16384 << 16;                               // tile_dim0 = 16384
        g1[4] = 1;                                         // tile_dim1 = 1
        g1[5] = 16384;                                     // tensor_dim0_stride
        v4i g2 = {};
        v4i g3 = {};
        v8i g4 = {};                                       // extra group (6-arg form)
        __builtin_amdgcn_tensor_load_to_lds(g0, g1, g2, g3, g4, 0);
    }

    const float* hr = h1  + (size_t)arow * HID_C;
    const float* ar = agg + (size_t)arow * HID_C;

    v8f acc[4] = {};
    const int nt0 = wave * 4;
    for (int kb = 0; kb < HID_C; kb += 4) {
        const int k  = kb + half * 2;
        const int kp = k >> 1;
        v2f aagg = *(const v2f*)(ar + k);
        v2f ah   = *(const v2f*)(hr + k);
#pragma unroll
        for (int t = 0; t < 4; ++t) {
            const int col = (nt0 + t) * 16 + l16;
            v2f bl = *(const v2f*)(W2lp + (size_t)kp * (2 * HID_C) + col * 2);
            v2f br = *(const v2f*)(W2rp + (size_t)kp * (2 * HID_C) + col * 2);
            acc[t] = __builtin_amdgcn_wmma_f32_16x16x4_f32(
                false, aagg, false, bl, (short)0, acc[t], false, false);
            acc[t] = __builtin_amdgcn_wmma_f32_16x16x4_f32(
                false, ah, false, br, (short)0, acc[t], false, false);
        }
    }
    // ReLU(h2) tile -> LDS
#pragma unroll
    for (int t = 0; t < 4; ++t) {
        const int col = (nt0 + t) * 16 + l16;
        const float bias = b2[col];
#pragma unroll
        for (int g = 0; g < 8; ++g) {
            const int r = g + half * 8;
            h2s[r][col] = fmaxf(acc[t][g] + bias, 0.0f);
        }
    }
    // Ensure the TDM transfer into Wa_s has landed, then sync the block.
    if (threadIdx.x < 32) __builtin_amdgcn_s_wait_tensorcnt(0);
    __syncthreads();

    // Head: logits[16x64]; wave w owns columns [16w, 16w+16). B from LDS.
    v8f acc2 = {};
    const int col = wave * 16 + l16;
    for (int kb = 0; kb < HID_C; kb += 4) {
        const int k  = kb + half * 2;
        const int kp = k >> 1;
        v2f a;
        a.x = h2s[l16][k];
        a.y = h2s[l16][k + 1];
        v2f b = *(const v2f*)(&Wa_s[kp * (2 * OUT_C) + col * 2]);
        acc2 = __builtin_amdgcn_wmma_f32_16x16x4_f32(
            false, a, false, b, (short)0, acc2, false, false);
    }
    const float bias = ba[col];
#pragma unroll
    for (int g = 0; g < 8; ++g) {
        const int r = rowbase + g + half * 8;
        out[(size_t)r * OUT_C + col] = acc2[g] + bias;
    }
}

// ---------------------------------------------------------------------------
extern "C" void kernel_launch(void* const* d_in, const int* in_sizes, int n_in,
                              void* d_out, int out_size, void* d_ws, size_t ws_size,
                              hipStream_t stream) {
    const float* x   = (const float*)d_in[0];
    const int*   ei  = (const int*)d_in[1];
    const float* W1l = (const float*)d_in[2];
    const float* b1  = (const float*)d_in[3];
    const float* W1r = (const float*)d_in[4];
    const float* W2l = (const float*)d_in[5];
    const float* b2  = (const float*)d_in[6];
    const float* W2r = (const float*)d_in[7];
    const float* Wa  = (const float*)d_in[8];
    const float* ba  = (const float*)d_in[9];
    float* out = (float*)d_out;

    // Workspace layout (4-byte units):
    //  deg[50048] | offs[50048] | cursor[50048] | csr[800000] |
    //  Wp[212992] | bufA[N*256] | bufB[N*256]
    int*   deg    = (int*)d_ws;
    int*   offs   = deg + 50048;
    int*   cursor = offs + 50048;
    int*   csr    = cursor + 50048;
    float* Wp     = (float*)(csr + N_EDGES);
    float* W1lp = Wp;
    float* W1rp = W1lp + IN_C * HID_C;       // 32768
    float* W2lp = W1rp + IN_C * HID_C;       // 32768
    float* W2rp = W2lp + HID_C * HID_C;      // 65536
    float* Wap  = W2rp + HID_C * HID_C;      // 65536
    float* bufA = Wap + HID_C * OUT_C;       // 16384; bufA = agg (both layers)
    float* bufB = bufA + (size_t)N_NODES * HID_C;   // h1

    const int row_blocks = N_NODES / 16;     // 3125, exact

    // --- CSR build (integer atomics only) ---
    zero_i32_kernel<<<(N_NODES + 255) / 256, 256, 0, stream>>>(deg, N_NODES);
    hist_kernel<<<N_EDGES / 256, 256, 0, stream>>>(ei, deg);
    scan_kernel<<<1, 1024, 0, stream>>>(deg, offs, cursor);
    scatter_edges_kernel<<<N_EDGES / 256, 256, 0, stream>>>(ei, cursor, csr);

    // --- Pair-pack all weight matrices for single-b64 B-fragment loads ---
    pack_pairs_kernel<<<(IN_C * HID_C + 255) / 256, 256, 0, stream>>>(W1l, W1lp, IN_C, HID_C);
    pack_pairs_kernel<<<(IN_C * HID_C + 255) / 256, 256, 0, stream>>>(W1r, W1rp, IN_C, HID_C);
    pack_pairs_kernel<<<(HID_C * HID_C + 255) / 256, 256, 0, stream>>>(W2l, W2lp, HID_C, HID_C);
    pack_pairs_kernel<<<(HID_C * HID_C + 255) / 256, 256, 0, stream>>>(W2r, W2rp, HID_C, HID_C);
    pack_pairs_kernel<<<(HID_C * OUT_C + 255) / 256, 256, 0, stream>>>(Wa, Wap, HID_C, OUT_C);

    // --- Layer 1 ---
    aggregate_kernel<IN_C><<<(N_NODES * 32) / 256, 256, 0, stream>>>(x, offs, csr, bufA);
    layer1_kernel<<<row_blocks, 128, 0, stream>>>(x, bufA, W1lp, b1, W1rp, bufB);

    // --- Layer 2 + head ---
    aggregate_kernel<HID_C><<<(N_NODES * 32) / 256, 256, 0, stream>>>(bufB, offs, csr, bufA);
    layer2_head_kernel<<<row_blocks, 128, 0, stream>>>(bufB, bufA, W2lp, b2, W2rp,
                                                       Wap, ba, out);
}